// MoleculeEncoder_60404420051621
// MI455X (gfx1250) — compile-verified
//
#include <hip/hip_runtime.h>
#include <hip/hip_bf16.h>

typedef __attribute__((ext_vector_type(2))) float v2f;
typedef __attribute__((ext_vector_type(8))) float v8f;

#define N_NODES 50000
#define HID     256
#define MTILES  (N_NODES / 16)   // 3125 exactly
#define BN_EPS  1e-5f

// ---------------------------------------------------------------- utilities

__global__ void zero_f32_kernel(float* __restrict__ p, int n) {
    int stride = gridDim.x * blockDim.x;
    for (int i = blockIdx.x * blockDim.x + threadIdx.x; i < n; i += stride)
        p[i] = 0.0f;
}

// ------------------------------------------------------------- edge scatter

// Layer-1 scatter: 9 features per node, one thread per edge.
__global__ void scatter9_kernel(const float* __restrict__ x,
                                const long long* __restrict__ src,
                                const long long* __restrict__ dst,
                                float* __restrict__ agg, int E) {
    int e = blockIdx.x * blockDim.x + threadIdx.x;
    if (e >= E) return;
    int s = (int)src[e];
    int d = (int)dst[e];
    #pragma unroll
    for (int f = 0; f < 9; ++f)
        unsafeAtomicAdd(&agg[d * 9 + f], x[s * 9 + f]);
}

// 256-feature scatter: one wave32 per edge; each lane owns 8 consecutive
// floats (two float4 loads), HW fp32 global atomics into the dst row.
__global__ __launch_bounds__(256) void scatter256_kernel(
        const float* __restrict__ h,
        const long long* __restrict__ src,
        const long long* __restrict__ dst,
        float* __restrict__ agg, int E) {
    int wave = threadIdx.x >> 5;
    int lane = threadIdx.x & 31;
    int e = blockIdx.x * 8 + wave;
    if (e >= E) return;
    int s = (int)src[e];
    int d = (int)dst[e];
    const float4* srow = (const float4*)(h + (size_t)s * HID);
    float4 v0 = srow[lane * 2 + 0];
    float4 v1 = srow[lane * 2 + 1];
    float* drow = agg + (size_t)d * HID + lane * 8;
    unsafeAtomicAdd(drow + 0, v0.x);
    unsafeAtomicAdd(drow + 1, v0.y);
    unsafeAtomicAdd(drow + 2, v0.z);
    unsafeAtomicAdd(drow + 3, v0.w);
    unsafeAtomicAdd(drow + 4, v1.x);
    unsafeAtomicAdd(drow + 5, v1.y);
    unsafeAtomicAdd(drow + 6, v1.z);
    unsafeAtomicAdd(drow + 7, v1.w);
}

// -------------------------------------------------- layer-1 MLP1 (K = 9)

__global__ __launch_bounds__(256) void l1_mlp1_kernel(
        const float* __restrict__ x, const float* __restrict__ agg9,
        const float* __restrict__ W1, const float* __restrict__ b1,
        float* __restrict__ out) {
    __shared__ float xa[9];
    int m = blockIdx.x;
    int n = threadIdx.x;
    if (n < 9) xa[n] = x[m * 9 + n] + agg9[m * 9 + n];
    __syncthreads();
    float s = b1[n];
    #pragma unroll
    for (int k = 0; k < 9; ++k)
        s = fmaf(xa[k], W1[k * HID + n], s);
    out[(size_t)m * HID + n] = s;
}

// ----------------------------------------------------- BatchNorm statistics

__global__ __launch_bounds__(256) void bn_stats_kernel(
        const float* __restrict__ h, float* __restrict__ stats) {
    int c = threadIdx.x;                  // channel 0..255
    float s = 0.f, ss = 0.f;
    for (int m = blockIdx.x; m < N_NODES; m += gridDim.x) {
        float v = h[(size_t)m * HID + c];
        s += v;
        ss += v * v;
    }
    unsafeAtomicAdd(&stats[c], s);
    unsafeAtomicAdd(&stats[HID + c], ss);
}

// scale = gamma * rsqrt(var + eps); shift = beta - mean * scale
__global__ void bn_finalize_kernel(float* __restrict__ stats,
                                   const float* __restrict__ gamma,
                                   const float* __restrict__ beta) {
    int c = threadIdx.x;
    float inv_n = 1.0f / (float)N_NODES;
    float mean = stats[c] * inv_n;
    float var  = stats[HID + c] * inv_n - mean * mean;
    float sc   = gamma[c] * rsqrtf(var + BN_EPS);
    stats[2 * HID + c] = sc;
    stats[3 * HID + c] = beta[c] - mean * sc;
}

// --------------------------------------------------------- WMMA f32 GEMM
// C[50000,256] = op_in(A)[50000,256] @ B[256,256] + bias, optional ReLU.
// One wave per 16x16 C tile, K looped in steps of 4 via v_wmma_f32_16x16x4_f32.
// A fragment (16x4 f32): lanes 0-15 -> rows M=0..15 with K=k0,k0+1;
//                        lanes 16-31 -> same rows with K=k0+2,k0+3.
// B fragment (4x16 f32): lanes 0-15 -> cols N=0..15 with K=k0,k0+1;
//                        lanes 16-31 -> same cols with K=k0+2,k0+3.
// C/D: VGPR i holds row (i + 8*(lane>=16)), col = lane&15.
// MODE_IN (compile-time): 0 = A; 1 = A + A2 (GIN x+agg);
//                         2 = relu(A*scale[k] + shift[k]) (BatchNorm).
template <int MODE_IN, int RELU_OUT>
__global__ __launch_bounds__(256) void gemm256_wmma_kernel(
        const float* __restrict__ A, const float* __restrict__ A2,
        const float* __restrict__ stats,          // scale @ +512, shift @ +768
        const float* __restrict__ B, const float* __restrict__ bias,
        float* __restrict__ out) {
    int lane = threadIdx.x & 31;
    int wave = threadIdx.x >> 5;
    int mt = blockIdx.x;                       // 0..3124
    int nt = blockIdx.y * 8 + wave;            // 0..15
    int row   = (mt << 4) + (lane & 15);
    int col   = (nt << 4) + (lane & 15);
    int khalf = (lane >> 4) << 1;              // 0 or 2

    v8f acc = {};

    // Branch-free, pointer-hoisted inner loop: all per-iteration addressing is
    // a constant offset, so loads clause and WMMAs issue back-to-back.
    const float* __restrict__ Arow  = A + (size_t)row * HID + khalf;
    const float* __restrict__ A2row = A2 + (size_t)row * HID + khalf;
    const float* __restrict__ Bcol  = B + col;
    const float* __restrict__ ssc   = stats + 2 * HID + khalf;
    const float* __restrict__ ssh   = stats + 3 * HID + khalf;

    #pragma unroll 8
    for (int k0 = 0; k0 < HID; k0 += 4) {
        if ((k0 & 31) == 0)
            __builtin_prefetch(Bcol + (size_t)(k0 + 32) * HID, 0, 1);
        float2 av = *(const float2*)(Arow + k0);
        if (MODE_IN == 1) {
            float2 g = *(const float2*)(A2row + k0);
            av.x += g.x;
            av.y += g.y;
        } else if (MODE_IN == 2) {
            float2 sc = *(const float2*)(ssc + k0);
            float2 sh = *(const float2*)(ssh + k0);
            av.x = fmaxf(fmaf(av.x, sc.x, sh.x), 0.0f);
            av.y = fmaxf(fmaf(av.y, sc.y, sh.y), 0.0f);
        }
        v2f a;
        a.x = av.x;
        a.y = av.y;
        v2f b;
        b.x = Bcol[(size_t)(k0 + khalf) * HID];
        b.y = Bcol[(size_t)(k0 + khalf + 1) * HID];
        acc = __builtin_amdgcn_wmma_f32_16x16x4_f32(
                  false, a, false, b, (short)0, acc, false, false);
    }

    float bv = bias[col];
    int rbase = (mt << 4) + ((lane >> 4) << 3);
    #pragma unroll
    for (int i = 0; i < 8; ++i) {
        float v = acc[i] + bv;
        if (RELU_OUT) v = fmaxf(v, 0.0f);
        out[(size_t)(rbase + i) * HID + col] = v;
    }
}

// ----------------------------------------------------------- final reduce

__global__ __launch_bounds__(256) void colsum_kernel(
        const float* __restrict__ h, float* __restrict__ out) {
    int c = threadIdx.x;
    float s = 0.f;
    for (int m = blockIdx.x; m < N_NODES; m += gridDim.x)
        s += h[(size_t)m * HID + c];
    unsafeAtomicAdd(&out[c], s);
}

// ---------------------------------------------------------------- driver

extern "C" void kernel_launch(void* const* d_in, const int* in_sizes, int n_in,
                              void* d_out, int out_size, void* d_ws, size_t ws_size,
                              hipStream_t stream) {
    const float* x = (const float*)d_in[0];
    const long long* ei = (const long long*)d_in[1];
    const int E = in_sizes[1] / 2;
    const long long* src = ei;
    const long long* dst = ei + E;

    // per-layer params: W1, b1, g, be, W2, b2 starting at d_in[2]
    const float* W1[3]; const float* b1[3]; const float* g[3];
    const float* be[3]; const float* W2[3]; const float* b2[3];
    for (int l = 0; l < 3; ++l) {
        W1[l] = (const float*)d_in[2 + 6 * l + 0];
        b1[l] = (const float*)d_in[2 + 6 * l + 1];
        g[l]  = (const float*)d_in[2 + 6 * l + 2];
        be[l] = (const float*)d_in[2 + 6 * l + 3];
        W2[l] = (const float*)d_in[2 + 6 * l + 4];
        b2[l] = (const float*)d_in[2 + 6 * l + 5];
    }

    const size_t NB = (size_t)N_NODES * HID;      // 12.8M floats per buffer
    float* P0 = (float*)d_ws;                      // GEMM1 output / BN input
    float* P1 = P0 + NB;                           // h_in / GEMM2 output
    float* P2 = P1 + NB;                           // agg
    float* stats = P2 + NB;                        // [sum|sumsq|scale|shift] = 1024 f32

    dim3 gemmGrid(MTILES, 2, 1);
    const int ZB = 2048;                           // zero-kernel blocks

    // ---- Layer 1 (F_in = 9) ----
    zero_f32_kernel<<<ZB, 256, 0, stream>>>(P1, N_NODES * 9);
    scatter9_kernel<<<(E + 255) / 256, 256, 0, stream>>>(x, src, dst, P1, E);
    l1_mlp1_kernel<<<N_NODES, 256, 0, stream>>>(x, P1, W1[0], b1[0], P0);
    zero_f32_kernel<<<4, 256, 0, stream>>>(stats, 1024);
    bn_stats_kernel<<<256, 256, 0, stream>>>(P0, stats);
    bn_finalize_kernel<<<1, 256, 0, stream>>>(stats, g[0], be[0]);
    gemm256_wmma_kernel<2, 1><<<gemmGrid, 256, 0, stream>>>(
        P0, nullptr, stats, W2[0], b2[0], P1);

    // ---- Layers 2 and 3 (256 -> 256) ----
    for (int l = 1; l < 3; ++l) {
        zero_f32_kernel<<<ZB, 256, 0, stream>>>(P2, (int)NB);
        scatter256_kernel<<<(E + 7) / 8, 256, 0, stream>>>(P1, src, dst, P2, E);
        gemm256_wmma_kernel<1, 0><<<gemmGrid, 256, 0, stream>>>(
            P1, P2, stats, W1[l], b1[l], P0);
        zero_f32_kernel<<<4, 256, 0, stream>>>(stats, 1024);
        bn_stats_kernel<<<256, 256, 0, stream>>>(P0, stats);
        bn_finalize_kernel<<<1, 256, 0, stream>>>(stats, g[l], be[l]);
        gemm256_wmma_kernel<2, 1><<<gemmGrid, 256, 0, stream>>>(
            P0, nullptr, stats, W2[l], b2[l], P1);
    }

    // ---- sum over nodes -> d_out[256] ----
    zero_f32_kernel<<<1, 256, 0, stream>>>((float*)d_out, out_size);
    colsum_kernel<<<256, 256, 0, stream>>>(P1, (float*)d_out);
}